// GIN_52956946760185
// MI455X (gfx1250) — compile-verified
//
#include <hip/hip_runtime.h>
#include <hip/hip_bf16.h>
#include <math.h>
#include <stdint.h>

// GIN on MI455X (gfx1250, wave32).
// - Edge aggregation: L2-resident scatter-add (global_atomic_add_f32).
// - All matmuls: V_WMMA_F32_16X16X4_F32 (exact fp32), BN folded into epilogue.
// - Weights staged per-block into LDS with gfx1250 async copy
//   (global_load_async_to_lds_b128 + s_wait_asynccnt), XOR-swizzled so the
//   B-fragment ds_load_b64 reads are bank-conflict-free on the 64-bank LDS.
// - log_softmax fused into the logits WMMA kernel via 16-lane shuffles.

typedef __attribute__((ext_vector_type(2))) float v2f;
typedef __attribute__((ext_vector_type(8))) float v8f;

#define BN_EPS 1e-5f

// ---------------- copy: dst = src (float4 vectorized) ----------------
__global__ void copy_f4(const float4* __restrict__ src, float4* __restrict__ dst, int n4) {
  int i = blockIdx.x * blockDim.x + threadIdx.x;
  if (i < n4) dst[i] = src[i];
}

// ------------- edge aggregation: agg[dst] += x[src] ------------------
// grid.y = 16 feature chunks of 4 floats; lanes cover consecutive edges so
// index loads are coalesced; gathers/atomics hit L2 (x is 25.6MB < 192MB L2).
__global__ void edge_agg(const float* __restrict__ x, const int* __restrict__ src,
                         const int* __restrict__ dst, float* __restrict__ agg, int E) {
  int e = blockIdx.x * blockDim.x + threadIdx.x;
  if (e >= E) return;
  int c = blockIdx.y * 4;
  int s = src[e], d = dst[e];
  const float4 v = *(const float4*)(x + (size_t)s * 64 + c);
  float* a = agg + (size_t)d * 64 + c;
  atomicAdd(a + 0, v.x);
  atomicAdd(a + 1, v.y);
  atomicAdd(a + 2, v.z);
  atomicAdd(a + 3, v.w);
}

// ---------------- WMMA GEMM: out = act(A @ W^T [+BN/bias]) ----------------
// A: [N x 64] row-major, W: [64 x 64] row-major (so B = W^T).
// One wave computes a 16-row x 64-col tile; 8 waves/block -> 128 rows/block.
// W is staged once per block into LDS via async copy; all 8 waves share it.
// MODE 0: BN (eval) folded affine + ReLU  (uses bias=b1, g, be, rm, rv)
// MODE 1: + bias, ReLU
// MODE 2: + bias, no ReLU
template<int MODE>
__global__ __launch_bounds__(256) void gemm_wmma_f32(
    const float* __restrict__ A, const float* __restrict__ W,
    const float* __restrict__ bias, const float* __restrict__ g,
    const float* __restrict__ be, const float* __restrict__ rm,
    const float* __restrict__ rv, float* __restrict__ out, int N)
{
  constexpr int K = 64;
  constexpr int NTILES = 4;
  constexpr int NOUT = 64;

  // ---- async-stage W into LDS, XOR-swizzled ----
  // W[n][k] lives at LDS dword  n*64 + (k ^ ((n&15)<<2)).
  // Mask only touches bits >=2, so each b128 chunk and each b64 fragment
  // pair stays contiguous; 32 lanes of a fragment read hit 32 disjoint
  // bank pairs (kb^4m / kb^4m+2) -> conflict-free.
  __shared__ float sW[4096];  // 16 KB
  {
    const int t = threadIdx.x;
#pragma unroll
    for (int j = 0; j < 4; ++j) {
      const int ci = t * 4 + j;          // float4 chunk id, 1024 total
      const int n = ci >> 4;             // weight row
      const int kc = ci & 15;            // float4 chunk within row
      const unsigned lds =
          (unsigned)(uintptr_t)&sW[(n * 16 + (kc ^ (n & 15))) * 4];
      const float* gp = W + (size_t)ci * 4;
      asm volatile("global_load_async_to_lds_b128 %0, %1, off"
                   :: "v"(lds), "v"(gp) : "memory");
    }
    asm volatile("s_wait_asynccnt 0x0" ::: "memory");
  }
  __syncthreads();

  const int lane  = threadIdx.x & 31;
  const int wave  = threadIdx.x >> 5;
  const int rowbase = (blockIdx.x * 8 + wave) * 16;
  const int m     = lane & 15;   // row (A) / col (B) within tile
  const int khalf = lane >> 4;   // K pair select per 16x4 f32 A layout
  int arow = rowbase + m;
  if (arow >= N) arow = N - 1;   // clamp loads; stores predicated (EXEC stays full)
  const float* Arow = A + (size_t)arow * K;
  const float* sWl  = sW + m * 64;   // this lane's B row base (swizzled)

  v8f acc[NTILES];
#pragma unroll
  for (int t = 0; t < NTILES; ++t) { v8f z = {}; acc[t] = z; }

#pragma unroll
  for (int kb = 0; kb < K; kb += 4) {
    // A frag: lane holds A[m][kb+2*khalf .. +1]  (16x4 f32 layout, 2 VGPRs)
    v2f a = *(const v2f*)(Arow + kb + 2 * khalf);
    const int bq = (kb + 2 * khalf) ^ (m << 2);  // swizzled dword offset
#pragma unroll
    for (int t = 0; t < NTILES; ++t) {
      // B frag: B[k][n] = W[n][k], n = t*16+m, from swizzled LDS
      v2f b = *(const v2f*)(sWl + t * 1024 + bq);
      acc[t] = __builtin_amdgcn_wmma_f32_16x16x4_f32(
          false, a, false, b, (short)0, acc[t], false, false);
    }
  }

#pragma unroll
  for (int t = 0; t < NTILES; ++t) {
    const int c = t * 16 + m;
    float alpha, beta;
    if (MODE == 0) {
      const float s = g[c] * rsqrtf(rv[c] + BN_EPS);
      alpha = s;
      beta  = (bias[c] - rm[c]) * s + be[c];
    } else {
      alpha = 1.0f;
      beta  = bias[c];
    }
#pragma unroll
    for (int i = 0; i < 8; ++i) {
      // C/D layout: VGPR i -> row i (lanes 0-15) / row i+8 (lanes 16-31)
      int r = rowbase + i + 8 * khalf;
      float v = acc[t][i] * alpha + beta;
      if (MODE != 2) v = fmaxf(v, 0.0f);
      if (r < N) out[(size_t)r * NOUT + c] = v;
    }
  }
}

// ------- final logits (NOUT=16) with fused log_softmax over 16 cols -------
__global__ __launch_bounds__(256) void logits_logsoftmax(
    const float* __restrict__ A, const float* __restrict__ W,
    const float* __restrict__ bias, float* __restrict__ out, int N)
{
  constexpr int K = 64;
  const int lane  = threadIdx.x & 31;
  const int wave  = threadIdx.x >> 5;
  const int rowbase = (blockIdx.x * 8 + wave) * 16;
  const int m     = lane & 15;
  const int khalf = lane >> 4;
  int arow = rowbase + m;
  if (arow >= N) arow = N - 1;
  const float* Arow = A + (size_t)arow * K;

  v8f acc = {};
#pragma unroll
  for (int kb = 0; kb < K; kb += 4) {
    v2f a = *(const v2f*)(Arow + kb + 2 * khalf);
    v2f b = *(const v2f*)(W + (size_t)m * K + kb + 2 * khalf);
    acc = __builtin_amdgcn_wmma_f32_16x16x4_f32(
        false, a, false, b, (short)0, acc, false, false);
  }

  const float bc = bias[m];
#pragma unroll
  for (int i = 0; i < 8; ++i) {
    int r = rowbase + i + 8 * khalf;
    float v = acc[i] + bc;
    // row r's 16 logits live across one 16-lane half -> reduce with shfl_xor
    float mx = v;
#pragma unroll
    for (int s = 8; s >= 1; s >>= 1) mx = fmaxf(mx, __shfl_xor(mx, s, 16));
    float e = expf(v - mx);
    float sum = e;
#pragma unroll
    for (int s = 8; s >= 1; s >>= 1) sum += __shfl_xor(sum, s, 16);
    float res = (v - mx) - logf(sum);
    if (r < N) out[(size_t)r * 16 + m] = res;
  }
}

// ---------------- global mean pool ----------------
__global__ void pool_accum(const float* __restrict__ h, const int* __restrict__ batch,
                           float* __restrict__ sums, float* __restrict__ cnts, int N) {
  int n = blockIdx.x * blockDim.x + threadIdx.x;
  if (n >= N) return;
  int c = blockIdx.y * 4;
  int gi = batch[n];
  const float4 v = *(const float4*)(h + (size_t)n * 64 + c);
  float* s = sums + (size_t)gi * 64 + c;
  atomicAdd(s + 0, v.x);
  atomicAdd(s + 1, v.y);
  atomicAdd(s + 2, v.z);
  atomicAdd(s + 3, v.w);
  if (blockIdx.y == 0) atomicAdd(&cnts[gi], 1.0f);
}

__global__ void pool_div(const float* __restrict__ sums, const float* __restrict__ cnts,
                         float* __restrict__ pooled, int total) {
  int i = blockIdx.x * blockDim.x + threadIdx.x;
  if (i >= total) return;
  pooled[i] = sums[i] / fmaxf(cnts[i >> 6], 1.0f);
}

extern "C" void kernel_launch(void* const* d_in, const int* in_sizes, int n_in,
                              void* d_out, int out_size, void* d_ws, size_t ws_size,
                              hipStream_t stream) {
  const float* x     = (const float*)d_in[0];
  const int*   edge  = (const int*)d_in[1];
  const int*   batch = (const int*)d_in[2];
  const int N = in_sizes[0] / 64;
  const int E = in_sizes[1] / 2;
  const int G = 512;
  const int* src = edge;       // edge_index[0]
  const int* dst = edge + E;   // edge_index[1]

  // params flattened: per layer {w1,b1,g,be,rm,rv,w2,b2}; layers conv1..3, lin
  auto LP = [&](int layer, int idx) { return (const float*)d_in[3 + layer * 8 + idx]; };

  // workspace: 3 ping-pong feature buffers + pool buffers (~77 MB)
  float* W0     = (float*)d_ws;                 // agg buffer   [N x 64]
  float* W1     = W0 + (size_t)N * 64;          // hidden       [N x 64]
  float* W2     = W1 + (size_t)N * 64;          // layer output [N x 64]
  float* sums   = W2 + (size_t)N * 64;          // [G x 64]
  float* cnts   = sums + (size_t)G * 64;        // [G]
  float* pooled = cnts + G;                     // [G x 64]
  float* hidden = pooled + (size_t)G * 64;      // [G x 64]

  const int nf4 = N * 64 / 4;
  dim3 cpg((nf4 + 255) / 256);
  dim3 eg((E + 255) / 256, 16);
  dim3 gg((N + 127) / 128);          // 128 rows per block (8 waves x 16)
  dim3 pg((N + 255) / 256, 16);

  const float* in = x;
  for (int layer = 0; layer < 3; ++layer) {
    // agg = in + segment_sum(in[src] -> dst)
    copy_f4<<<cpg, 256, 0, stream>>>((const float4*)in, (float4*)W0, nf4);
    edge_agg<<<eg, 256, 0, stream>>>(in, src, dst, W0, E);
    // Linear1 + BN + ReLU
    gemm_wmma_f32<0><<<gg, 256, 0, stream>>>(
        W0, LP(layer, 0), LP(layer, 1), LP(layer, 2), LP(layer, 3),
        LP(layer, 4), LP(layer, 5), W1, N);
    // Linear2 (+ outer ReLU for conv1/conv2)
    if (layer < 2)
      gemm_wmma_f32<1><<<gg, 256, 0, stream>>>(
          W1, LP(layer, 6), LP(layer, 7), nullptr, nullptr, nullptr, nullptr, W2, N);
    else
      gemm_wmma_f32<2><<<gg, 256, 0, stream>>>(
          W1, LP(layer, 6), LP(layer, 7), nullptr, nullptr, nullptr, nullptr, W2, N);
    in = W2;
  }

  // global mean pool
  hipMemsetAsync(sums, 0, (size_t)(G * 64 + G) * sizeof(float), stream);
  pool_accum<<<pg, 256, 0, stream>>>(W2, batch, sums, cnts, N);
  pool_div<<<(G * 64 + 255) / 256, 256, 0, stream>>>(sums, cnts, pooled, G * 64);

  // classifier: Linear + BN + ReLU, then Linear(16) + log_softmax
  dim3 lg((G + 127) / 128);
  gemm_wmma_f32<0><<<lg, 256, 0, stream>>>(
      pooled, LP(3, 0), LP(3, 1), LP(3, 2), LP(3, 3), LP(3, 4), LP(3, 5), hidden, G);
  logits_logsoftmax<<<lg, 256, 0, stream>>>(hidden, LP(3, 6), LP(3, 7), (float*)d_out, G);
}